// QueryAndGroup_23287312679504
// MI455X (gfx1250) — compile-verified
//
#include <hip/hip_runtime.h>
#include <cstdint>

typedef float v2f __attribute__((ext_vector_type(2)));
typedef float v8f __attribute__((ext_vector_type(8)));
typedef int   v4i_vs __attribute__((vector_size(16)));   // matches builtin's pointee type

#if defined(__has_builtin)
#if __has_builtin(__builtin_amdgcn_global_load_async_to_lds_b128)
#define HAVE_ASYNC 1
#endif
#endif
#ifndef HAVE_ASYNC
#define HAVE_ASYNC 0
#endif

namespace {
constexpr int BB = 4;
constexpr int NN = 16384;
constexpr int SS = 4096;
constexpr int CC = 64;
constexpr int KK = 64;            // nsample
constexpr float R2 = 0.04f;       // radius^2 = 0.2^2
constexpr int OC = 3 + CC;        // 67 output channels

constexpr int QPW   = 16;         // queries per wave (WMMA M)
constexpr int WAVES = 2;          // waves per block
constexpr int TPB   = WAVES * 32; // 64 threads
constexpr int QPB   = QPW * WAVES;
constexpr int TILE  = 16;         // points per WMMA tile
constexpr int CTILES = 128;       // tiles per chunk
constexpr int CHUNK  = CTILES * TILE;   // 2048 points
constexpr int NCHUNKS = NN / CHUNK;     // 8
constexpr int NBUF   = HAVE_ASYNC ? 2 : 1;
constexpr int VEC4S  = CHUNK * 3 / 4;   // 1536 uint4 per chunk
constexpr int V4PT   = VEC4S / TPB;     // 24 uint4 per thread
}

// Deposit a uniform (wave-scalar) value into lane `r` of vtmp.
static __device__ __forceinline__ uint32_t lane_deposit(uint32_t vtmp, uint32_t val,
                                                        int r, int lane) {
#if __has_builtin(__builtin_amdgcn_writelane)
    return __builtin_amdgcn_writelane(val, r, vtmp);
#else
    return (lane == r) ? val : vtmp;   // val is uniform -> v_cmp + v_cndmask (masks hoisted)
#endif
}

static __device__ __forceinline__ void wait_async0() {
#if __has_builtin(__builtin_amdgcn_s_wait_asynccnt)
    __builtin_amdgcn_s_wait_asynccnt(0);
#else
    __asm__ volatile("s_wait_asynccnt 0x0" ::: "memory");
#endif
}

// Cooperatively stage one chunk (24 KB of xyz) into LDS.
static __device__ __forceinline__ void stage_chunk(const float* __restrict__ src,
                                                   float* dst_lds, int tid) {
    const uint4* __restrict__ g = (const uint4*)src;
    uint4* l = (uint4*)dst_lds;
#if HAVE_ASYNC
#pragma unroll
    for (int k = 0; k < V4PT; ++k) {
        const int i = k * TPB + tid;
        __builtin_amdgcn_global_load_async_to_lds_b128(
            (__attribute__((address_space(1))) v4i_vs*)(g + i),
            (__attribute__((address_space(3))) v4i_vs*)(l + i),
            0, 0);
    }
#else
#pragma unroll
    for (int k = 0; k < V4PT; k += 4) {   // 4 loads in flight, then 4 stores
        uint4 t0 = g[(k+0)*TPB + tid];
        uint4 t1 = g[(k+1)*TPB + tid];
        uint4 t2 = g[(k+2)*TPB + tid];
        uint4 t3 = g[(k+3)*TPB + tid];
        l[(k+0)*TPB + tid] = t0;
        l[(k+1)*TPB + tid] = t1;
        l[(k+2)*TPB + tid] = t2;
        l[(k+3)*TPB + tid] = t3;
    }
#endif
}

__global__ __launch_bounds__(TPB)
void ballquery_group_kernel(const float* __restrict__ xyz,
                            const float* __restrict__ new_xyz,
                            const float* __restrict__ feat,
                            float* __restrict__ out)
{
    __shared__ float    ptsbuf[NBUF][CHUNK * 3];   // staged xyz chunk(s)
    __shared__ uint32_t maskw[WAVES][8][CTILES];   // 8 KB: per-tile ballot dwords
    __shared__ uint16_t idxl[WAVES][QPW][KK];      // 4 KB: per-query index lists
    __shared__ int      donef[WAVES];

    const int tid  = threadIdx.x;
    const int lane = tid & 31;
    const int wave = tid >> 5;
    const int m    = lane & 15;
    const bool hi  = lane >= 16;

    const int b     = blockIdx.y;
    const int sbase = blockIdx.x * QPB + wave * QPW;

    const float* __restrict__ xyzb = xyz     + (size_t)b * NN * 3;
    const float* __restrict__ nxb  = new_xyz + (size_t)b * SS * 3;
    const float* __restrict__ fb   = feat    + (size_t)b * CC * NN;

    // ---- per-wave setup: A (16x4) and C (|q|^2) matrices, constant over N ----
    const int sq = sbase + m;
    const float qx = nxb[sq*3+0];
    const float qy = nxb[sq*3+1];
    const float qz = nxb[sq*3+2];
    const float q2 = qx*qx + qy*qy + qz*qz;

    // A layout (16x4 f32): lanes0-15 hold K=0,1 in v0,v1; lanes16-31 K=2,3
    v2f amat;
    amat.x = hi ? (-2.0f*qz) : (-2.0f*qx);
    amat.y = hi ? 1.0f       : (-2.0f*qy);

    // C layout: VGPR r = row r (lanes0-15) / row r+8 (lanes16-31); broadcast q2[row]
    v8f cmat;
    const int selbase = hi ? 32 : 0;
#pragma unroll
    for (int r = 0; r < 8; ++r) {
        int v = __builtin_amdgcn_ds_bpermute(selbase + r*4, __float_as_int(q2));
        cmat[r] = __int_as_float(v);
    }

    int cnt = hi ? KK : 0;      // hi lanes "done" so full-wave ballots work
    bool wavedone = false;      // wave-uniform

#if HAVE_ASYNC
    stage_chunk(xyzb, ptsbuf[0], tid);      // kick chunk 0
#endif
    int buf = 0;
    for (int chnk = 0; chnk < NCHUNKS; ++chnk) {
        const int nbase = chnk * CHUNK;
#if HAVE_ASYNC
        wait_async0();
        __syncthreads();        // all waves' staging of ptsbuf[buf] complete
        if (chnk + 1 < NCHUNKS) // stream next chunk into the other buffer
            stage_chunk(xyzb + (size_t)(nbase + CHUNK) * 3, ptsbuf[buf ^ 1], tid);
#else
        __syncthreads();        // prior reads of ptsbuf done
        stage_chunk(xyzb + (size_t)nbase * 3, ptsbuf[0], tid);
        __syncthreads();        // staging visible
#endif

        if (!wavedone) {
            const float* __restrict__ pbuf = ptsbuf[buf];

            // ---- phase 1: WMMA distance tiles -> ballot masks -> LDS ----
            for (int t = 0; t < CTILES; ++t) {
                const float* pp = pbuf + ((size_t)t * TILE + m) * 3;
                const float cx = pp[0], cy = pp[1], cz = pp[2];
                const float p2 = cx*cx + cy*cy + cz*cz;
                // B layout (4x16 f32): lanes0-15 K=0,1 in v0,v1; lanes16-31 K=2,3
                v2f bmat;
                bmat.x = hi ? cz : cx;
                bmat.y = hi ? p2 : cy;
                // d2[row][col] = q2 + p2 - 2 q.p
                v8f d = __builtin_amdgcn_wmma_f32_16x16x4_f32(
                    false, amat, false, bmat, (short)0, cmat, false, false);
                uint32_t vtmp = 0;
#pragma unroll
                for (int r = 0; r < 8; ++r) {
                    uint32_t bal = __builtin_amdgcn_ballot_w32(d[r] < R2);
                    vtmp = lane_deposit(vtmp, bal, r, lane);
                }
                if (lane < 8) maskw[wave][lane][t] = vtmp;
            }
            __asm__ volatile("" ::: "memory");  // same-wave LDS is HW-ordered

            // ---- phase 2: per-query first-64 compaction (ascending order) ----
            if (!hi) {
                const int r8 = m & 7;
                const int sh = (m >> 3) * 16;
                for (int t = 0; t < CTILES; ++t) {
                    if (__builtin_amdgcn_ballot_w32(cnt < KK) == 0u) break;
                    uint32_t w  = maskw[wave][r8][t];
                    uint32_t mk = (w >> sh) & 0xFFFFu;
                    const int n0 = nbase + t * TILE;
                    while (mk != 0u && cnt < KK) {
                        const int bit = __builtin_ctz(mk);
                        idxl[wave][m][cnt] = (uint16_t)(n0 + bit);
                        ++cnt;
                        mk &= mk - 1u;
                    }
                }
            }
            wavedone = (__builtin_amdgcn_ballot_w32(cnt < KK) == 0u);
        }

        if (lane == 0) donef[wave] = wavedone ? 1 : 0;
        __syncthreads();                    // donef visible; reads of ptsbuf[buf] done
        if (donef[0] && donef[1]) break;    // block-uniform exit
#if HAVE_ASYNC
        buf ^= 1;
#endif
    }

    // ---- pad lists: repeat first index, or 0 if none found ----
    if (!hi) {
        const uint16_t first = (cnt > 0) ? idxl[wave][m][0] : (uint16_t)0;
        for (int j = cnt; j < KK; ++j) idxl[wave][m][j] = first;
    }
    __asm__ volatile("" ::: "memory");

    // ---- phase 3: gather + coalesced writes; lane owns k = 2*lane, 2*lane+1 ----
    const size_t chstride = (size_t)SS * KK;
    float* __restrict__ outb = out + (size_t)b * OC * chstride;
    for (int jq = 0; jq < QPW; ++jq) {
        const int s = sbase + jq;
        const uint32_t w = ((const uint32_t*)idxl[wave][jq])[lane];
        const int n0 = (int)(w & 0xFFFFu);
        const int n1 = (int)(w >> 16);

        const float* p0 = xyzb + (size_t)n0 * 3;
        const float* p1 = xyzb + (size_t)n1 * 3;
        const float x0 = p0[0], y0 = p0[1], z0 = p0[2];
        const float x1 = p1[0], y1 = p1[1], z1 = p1[2];
        const float gx = nxb[s*3+0], gy = nxb[s*3+1], gz = nxb[s*3+2];

        float* op = outb + (size_t)s * KK + 2*lane;
        ((float2*)(op + 0*chstride))[0] = make_float2(x0-gx, x1-gx);
        ((float2*)(op + 1*chstride))[0] = make_float2(y0-gy, y1-gy);
        ((float2*)(op + 2*chstride))[0] = make_float2(z0-gz, z1-gz);

        const float* f0 = fb + n0;
        const float* f1 = fb + n1;
        float* oc = op + 3*chstride;
#pragma unroll 8
        for (int c = 0; c < CC; ++c) {
            ((float2*)oc)[0] = make_float2(f0[(size_t)c*NN], f1[(size_t)c*NN]);
            oc += chstride;
        }
    }
}

extern "C" void kernel_launch(void* const* d_in, const int* in_sizes, int n_in,
                              void* d_out, int out_size, void* d_ws, size_t ws_size,
                              hipStream_t stream) {
    (void)in_sizes; (void)n_in; (void)out_size; (void)d_ws; (void)ws_size;
    const float* xyz     = (const float*)d_in[0];
    const float* new_xyz = (const float*)d_in[1];
    const float* feat    = (const float*)d_in[2];
    float* out           = (float*)d_out;

    dim3 grid(SS / QPB, BB);      // (128, 4) blocks
    dim3 block(TPB);              // 64 threads = 2 waves
    hipLaunchKernelGGL(ballquery_group_kernel, grid, block, 0, stream,
                       xyz, new_xyz, feat, out);
}